// LOTCModel_24017457119868
// MI455X (gfx1250) — compile-verified
//
#include <hip/hip_runtime.h>
#include <hip/hip_bf16.h>
#include <math.h>

typedef float v2f __attribute__((ext_vector_type(2)));
typedef float v8f __attribute__((ext_vector_type(8)));

#define NB      8192     // batch
#define DIN     512
#define DH      256
#define NK      1024
#define EPS_OT  0.05f
#define INV_EPS 20.0f
#define LOG_U   (-9.010913347279289f)   // -log(8192)
#define NCHUNK  32                      // row chunks for column LSE

// ---------------- block reduction helpers (blockDim.x == 256) ----------------
__device__ inline float blockReduceSum256(float v, float* sd) {
  int t = threadIdx.x;
  sd[t] = v; __syncthreads();
  for (int s = 128; s > 0; s >>= 1) { if (t < s) sd[t] += sd[t + s]; __syncthreads(); }
  float r = sd[0]; __syncthreads();
  return r;
}
__device__ inline float blockReduceMax256(float v, float* sd) {
  int t = threadIdx.x;
  sd[t] = v; __syncthreads();
  for (int s = 128; s > 0; s >>= 1) { if (t < s) sd[t] = fmaxf(sd[t], sd[t + s]); __syncthreads(); }
  float r = sd[0]; __syncthreads();
  return r;
}

// ---------------- init: zero g + accumulators ----------------
__global__ __launch_bounds__(256)
void init_kernel(float* g, float* ot_acc, float* disp_acc) {
  int t = blockIdx.x * 256 + threadIdx.x;
  if (t < NK) g[t] = 0.0f;
  if (t == 0) { *ot_acc = 0.0f; *disp_acc = 0.0f; }
}

// ---------------- encode GEMM: out = x @ W + b  (fp32 WMMA 16x16x4) ----------------
// one wave per 16x16 tile. A layout (16x4 f32): lanes 0-15 hold K=0,1 in v0,v1;
// lanes 16-31 hold K=2,3. B mirrors (K rows striped the same way).
__global__ __launch_bounds__(32)
void encode_gemm(const float* __restrict__ x, const float* __restrict__ W,
                 const float* __restrict__ bias, float* __restrict__ out) {
  const int lane = threadIdx.x;
  const int col0 = blockIdx.x * 16;
  const int row0 = blockIdx.y * 16;
  const int m  = lane & 15;
  const int kh = lane >> 4;
  v8f acc = {};
  const float* xrow = x + (size_t)(row0 + m) * DIN;
  for (int k0 = 0; k0 < DIN; k0 += 4) {
    const int ka = k0 + 2 * kh;
    v2f a, b;
    a.x = xrow[ka];
    a.y = xrow[ka + 1];
    b.x = W[(size_t)ka * DH + col0 + m];
    b.y = W[(size_t)(ka + 1) * DH + col0 + m];
    acc = __builtin_amdgcn_wmma_f32_16x16x4_f32(false, a, false, b, (short)0, acc,
                                                false, false);
  }
  const float bv = bias[col0 + m];
#pragma unroll
  for (int v = 0; v < 8; ++v) {
    const int row = row0 + v + 8 * kh;           // C/D layout: lanes16-31 -> M+8
    out[(size_t)row * DH + col0 + m] = acc[v] + bv;
  }
}

// ---------------- cost GEMM: Cmat = 1 - z @ c^T  (fp32 WMMA) ----------------
__global__ __launch_bounds__(32)
void cost_gemm(const float* __restrict__ z, const float* __restrict__ c,
               float* __restrict__ Cmat) {
  const int lane = threadIdx.x;
  const int col0 = blockIdx.x * 16;   // over K=1024 prototypes
  const int row0 = blockIdx.y * 16;   // over B=8192 samples
  const int m  = lane & 15;
  const int kh = lane >> 4;
  v8f acc = {};
  const float* zrow = z + (size_t)(row0 + m) * DH;
  const float* crow = c + (size_t)(col0 + m) * DH;   // B^T row = prototype row
  for (int k0 = 0; k0 < DH; k0 += 4) {
    const int ka = k0 + 2 * kh;
    v2f a, b;
    a.x = zrow[ka]; a.y = zrow[ka + 1];
    b.x = crow[ka]; b.y = crow[ka + 1];
    acc = __builtin_amdgcn_wmma_f32_16x16x4_f32(false, a, false, b, (short)0, acc,
                                                false, false);
  }
#pragma unroll
  for (int v = 0; v < 8; ++v) {
    const int row = row0 + v + 8 * kh;
    Cmat[(size_t)row * NK + col0 + m] = 1.0f - acc[v];
  }
}

// ---------------- dispersion: sum_{i!=j} exp(2*dot(ci,cj) - 2) via WMMA Gram ----------------
__global__ __launch_bounds__(32)
void disp_gemm(const float* __restrict__ c, float* __restrict__ disp_acc) {
  const int lane = threadIdx.x;
  const int col0 = blockIdx.x * 16;
  const int row0 = blockIdx.y * 16;
  const int m  = lane & 15;
  const int kh = lane >> 4;
  v8f acc = {};
  const float* arow = c + (size_t)(row0 + m) * DH;
  const float* brow = c + (size_t)(col0 + m) * DH;
  for (int k0 = 0; k0 < DH; k0 += 4) {
    const int ka = k0 + 2 * kh;
    v2f a, b;
    a.x = arow[ka]; a.y = arow[ka + 1];
    b.x = brow[ka]; b.y = brow[ka + 1];
    acc = __builtin_amdgcn_wmma_f32_16x16x4_f32(false, a, false, b, (short)0, acc,
                                                false, false);
  }
  float s = 0.0f;
#pragma unroll
  for (int v = 0; v < 8; ++v) {
    const int row = row0 + v + 8 * kh;
    const int col = col0 + m;
    if (row != col) s += __expf(2.0f * acc[v] - 2.0f);
  }
  for (int off = 16; off > 0; off >>= 1) s += __shfl_down(s, off, 32);
  if (lane == 0) atomicAdd(disp_acc, s);
}

// ---------------- row L2 normalization (256-wide rows; may run in place) ----------------
__global__ __launch_bounds__(256)
void row_l2_normalize(const float* __restrict__ in, float* __restrict__ out) {
  __shared__ float sd[256];
  const int row = blockIdx.x;
  const int t = threadIdx.x;
  const float v = in[(size_t)row * DH + t];
  const float ss = blockReduceSum256(v * v, sd);
  const float nrm = fmaxf(sqrtf(ss), 1e-12f);
  out[(size_t)row * DH + t] = v / nrm;
}

// ---------------- masses = softmax(mass_logits); log_v; reg_mass ----------------
__global__ __launch_bounds__(1024)
void mass_softmax(const float* __restrict__ logits, float* __restrict__ masses,
                  float* __restrict__ logv, float* __restrict__ reg_mass) {
  __shared__ float sd[1024];
  const int t = threadIdx.x;
  const float v = logits[t];
  sd[t] = v; __syncthreads();
  for (int s = 512; s > 0; s >>= 1) { if (t < s) sd[t] = fmaxf(sd[t], sd[t + s]); __syncthreads(); }
  const float mx = sd[0]; __syncthreads();
  const float e = __expf(v - mx);
  sd[t] = e; __syncthreads();
  for (int s = 512; s > 0; s >>= 1) { if (t < s) sd[t] += sd[t + s]; __syncthreads(); }
  const float sum = sd[0]; __syncthreads();
  const float mval = e / sum;
  masses[t] = mval;
  const float lv = __logf(mval + 1e-12f);
  logv[t] = lv;
  sd[t] = mval * lv; __syncthreads();
  for (int s = 512; s > 0; s >>= 1) { if (t < s) sd[t] += sd[t + s]; __syncthreads(); }
  if (t == 0) *reg_mass = sd[0];
}

// ---------------- Sinkhorn row pass: f[i] = eps*(log_u - LSE_k((g-C)/eps)) ----------------
__global__ __launch_bounds__(256)
void sinkhorn_row(const float* __restrict__ Cmat, const float* __restrict__ g,
                  float* __restrict__ f) {
  __shared__ float sd[256];
  const int i = blockIdx.x;
  const int t = threadIdx.x;
  const float* crow = Cmat + (size_t)i * NK;
  float v[4];
  float mx = -INFINITY;
#pragma unroll
  for (int j = 0; j < 4; ++j) {
    const int k = t + 256 * j;
    v[j] = (g[k] - crow[k]) * INV_EPS;
    mx = fmaxf(mx, v[j]);
  }
  mx = blockReduceMax256(mx, sd);
  float s = 0.0f;
#pragma unroll
  for (int j = 0; j < 4; ++j) s += __expf(v[j] - mx);
  s = blockReduceSum256(s, sd);
  if (t == 0) f[i] = EPS_OT * (LOG_U - (mx + __logf(s)));
}

// ---------------- Sinkhorn column pass, stage 1: online LSE per (chunk, k) ----------------
__global__ __launch_bounds__(256)
void sinkhorn_col_part(const float* __restrict__ Cmat, const float* __restrict__ f,
                       float* __restrict__ partm, float* __restrict__ parts) {
  __shared__ float sf[256];
  const int k = blockIdx.x * 256 + threadIdx.x;
  const int i0 = blockIdx.y * 256;
  sf[threadIdx.x] = f[i0 + threadIdx.x];
  __syncthreads();
  float m = -INFINITY, s = 0.0f;
  for (int ii = 0; ii < 256; ++ii) {
    const float v = (sf[ii] - Cmat[(size_t)(i0 + ii) * NK + k]) * INV_EPS;
    const float nm = fmaxf(m, v);
    s = s * __expf(m - nm) + __expf(v - nm);
    m = nm;
  }
  partm[(size_t)blockIdx.y * NK + k] = m;
  parts[(size_t)blockIdx.y * NK + k] = s;
}

// ---------------- Sinkhorn column pass, stage 2: combine + g update ----------------
__global__ __launch_bounds__(256)
void sinkhorn_col_combine(const float* __restrict__ partm, const float* __restrict__ parts,
                          const float* __restrict__ logv, float* __restrict__ g) {
  const int k = blockIdx.x * 256 + threadIdx.x;
  float m = -INFINITY, s = 0.0f;
  for (int c = 0; c < NCHUNK; ++c) {
    const float pm = partm[(size_t)c * NK + k];
    const float ps = parts[(size_t)c * NK + k];
    const float nm = fmaxf(m, pm);
    s = s * __expf(m - nm) + ps * __expf(pm - nm);
    m = nm;
  }
  g[k] = EPS_OT * (logv[k] - (m + __logf(s)));
}

// ---------------- final: P, ot_cost partial, soft, hard ----------------
__global__ __launch_bounds__(256)
void finalize_P(const float* __restrict__ Cmat, const float* __restrict__ f,
                const float* __restrict__ g, float* __restrict__ P_out,
                float* __restrict__ soft_out, float* __restrict__ hard_out,
                float* __restrict__ ot_acc) {
  __shared__ float sd[256];
  __shared__ float sv[256];
  __shared__ int   si[256];
  const int i = blockIdx.x;
  const int t = threadIdx.x;
  const float fi = f[i];
  const float* crow = Cmat + (size_t)i * NK;
  float p[4];
  float rs = 0.0f, ot = 0.0f, bestv = -INFINITY;
  int bestk = 0;
#pragma unroll
  for (int j = 0; j < 4; ++j) {
    const int k = t + 256 * j;
    const float cv = crow[k];
    const float pv = __expf((fi + g[k] - cv) * INV_EPS);
    p[j] = pv;
    rs += pv;
    ot += pv * cv;
    if (pv > bestv) { bestv = pv; bestk = k; }
    P_out[(size_t)i * NK + k] = pv;
  }
  rs = blockReduceSum256(rs, sd);
  ot = blockReduceSum256(ot, sd);
  sv[t] = bestv; si[t] = bestk; __syncthreads();
  for (int s = 128; s > 0; s >>= 1) {
    if (t < s) {
      const float ov = sv[t + s]; const int oi = si[t + s];
      if (ov > sv[t] || (ov == sv[t] && oi < si[t])) { sv[t] = ov; si[t] = oi; }
    }
    __syncthreads();
  }
  if (t == 0) {
    hard_out[i] = (float)si[0];
    atomicAdd(ot_acc, ot);
  }
  const float inv_rs = 1.0f / (rs + 1e-12f);
#pragma unroll
  for (int j = 0; j < 4; ++j)
    soft_out[(size_t)i * NK + t + 256 * j] = p[j] * inv_rs;
}

// ---------------- scalar finalize ----------------
__global__ void finalize_scalars(const float* __restrict__ ot, const float* __restrict__ mass,
                                 const float* __restrict__ disp, float* __restrict__ scal) {
  const float o = *ot;
  const float m = *mass;
  const float d = *disp / ((float)NK * (float)(NK - 1));  // off-diagonal already excluded
  scal[0] = o;
  scal[1] = m;
  scal[2] = d;
  scal[3] = o + 0.1f * m + 0.1f * d;
}

extern "C" void kernel_launch(void* const* d_in, const int* in_sizes, int n_in,
                              void* d_out, int out_size, void* d_ws, size_t ws_size,
                              hipStream_t stream) {
  const float* x     = (const float*)d_in[0];
  const float* W     = (const float*)d_in[1];
  const float* b     = (const float*)d_in[2];
  const float* proto = (const float*)d_in[3];
  const float* mlog  = (const float*)d_in[4];

  float* out   = (float*)d_out;
  float* z_out = out;                                   // 8192*256
  float* P_out = out + (size_t)NB * DH;                 // 8192*1024
  float* scal  = P_out + (size_t)NB * NK;               // 4 scalars
  float* hard  = scal + 4;                              // 8192
  float* soft  = hard + NB;                             // 8192*1024

  float* ws       = (float*)d_ws;
  float* c_norm   = ws;                                 // 1024*256
  float* Cmat     = c_norm + (size_t)NK * DH;           // 8192*1024 (L2-resident)
  float* masses   = Cmat + (size_t)NB * NK;             // 1024
  float* logv     = masses + NK;                        // 1024
  float* fvec     = logv + NK;                          // 8192
  float* gvec     = fvec + NB;                          // 1024
  float* partm    = gvec + NK;                          // 32*1024
  float* parts    = partm + (size_t)NCHUNK * NK;        // 32*1024
  float* ot_acc   = parts + (size_t)NCHUNK * NK;        // 1
  float* disp_acc = ot_acc + 1;                         // 1
  float* mass_acc = disp_acc + 1;                       // 1

  init_kernel<<<4, 256, 0, stream>>>(gvec, ot_acc, disp_acc);

  encode_gemm<<<dim3(DH / 16, NB / 16), 32, 0, stream>>>(x, W, b, z_out);
  row_l2_normalize<<<NB, 256, 0, stream>>>(z_out, z_out);       // in place
  row_l2_normalize<<<NK, 256, 0, stream>>>(proto, c_norm);
  mass_softmax<<<1, 1024, 0, stream>>>(mlog, masses, logv, mass_acc);

  cost_gemm<<<dim3(NK / 16, NB / 16), 32, 0, stream>>>(z_out, c_norm, Cmat);
  disp_gemm<<<dim3(NK / 16, NK / 16), 32, 0, stream>>>(c_norm, disp_acc);

  for (int it = 0; it < 50; ++it) {
    sinkhorn_row<<<NB, 256, 0, stream>>>(Cmat, gvec, fvec);
    sinkhorn_col_part<<<dim3(NK / 256, NCHUNK), 256, 0, stream>>>(Cmat, fvec, partm, parts);
    sinkhorn_col_combine<<<NK / 256, 256, 0, stream>>>(partm, parts, logv, gvec);
  }

  finalize_P<<<NB, 256, 0, stream>>>(Cmat, fvec, gvec, P_out, soft, hard, ot_acc);
  finalize_scalars<<<1, 1, 0, stream>>>(ot_acc, mass_acc, disp_acc, scal);
}